// LightGCN_40063454937773
// MI455X (gfx1250) — compile-verified
//
#include <hip/hip_runtime.h>
#include <hip/hip_bf16.h>

// ---------------------------------------------------------------------------
// LightGCN propagation on MI455X (gfx1250).
// Memory-bound sparse gather/scatter; key gfx1250 paths used:
//   * global_load_async_to_lds_b128 (CDNA5 async copy) to stage edge tiles
//   * s_wait_asynccnt for async completion
//   * non-temporal (TH=NT) b128 streams for the touch-once accumulate passes
//   * wave32-sized 256-thread blocks, b128 vector memory ops
//   * hardware f32 global atomics for the scatter-add
// ---------------------------------------------------------------------------

#define DIM            64
#define TPB            256
#define LANES_PER_EDGE 16                       // 16 lanes x float4 = 64 floats
#define EDGES_INFLIGHT (TPB / LANES_PER_EDGE)   // 16 edges at a time
#define EDGE_TILE      256                      // edges staged per block

#if defined(__HIP_DEVICE_COMPILE__) && defined(__gfx1250__)
#define GFX1250_DEV 1
#endif

typedef int   v4i_t __attribute__((ext_vector_type(4)));
typedef float v4f_t __attribute__((ext_vector_type(4)));

// --- CDNA5 async global->LDS copy (16B per lane), with safe fallbacks -------
// Builtin signature (verified by hipcc): (v4i32 AS1*, v4i32 AS3*, imm, imm)
__device__ __forceinline__ void tile_copy_b128(const void* gsrc, void* ldst) {
#if defined(GFX1250_DEV) && __has_builtin(__builtin_amdgcn_global_load_async_to_lds_b128)
    void* g = const_cast<void*>(gsrc);
    __builtin_amdgcn_global_load_async_to_lds_b128(
        (__attribute__((address_space(1))) v4i_t*)g,
        (__attribute__((address_space(3))) v4i_t*)ldst,
        /*offset=*/0, /*cpol=*/0);
#else
    // Synchronous fallback (still correct; no ASYNCcnt traffic)
    *reinterpret_cast<int4*>(ldst) = *reinterpret_cast<const int4*>(gsrc);
#endif
}

__device__ __forceinline__ void wait_async_copies() {
#if defined(GFX1250_DEV)
#if __has_builtin(__builtin_amdgcn_s_wait_asynccnt)
    __builtin_amdgcn_s_wait_asynccnt(0);
#else
    asm volatile("s_wait_asynccnt 0" ::: "memory");
#endif
#endif
}

// ---------------------------------------------------------------------------
// scatter: xn[col[e]] += w[e] * x[row[e]]   (one layer of SpMM)
// Block stages EDGE_TILE edges (row/col/w) into LDS with async b128 copies,
// then 16 lanes per edge each gather a float4 of x[row] and do 4 f32 atomics.
// (Placed first in the file so the disasm snippet shows this kernel.)
// ---------------------------------------------------------------------------
__global__ __launch_bounds__(TPB) void lightgcn_scatter(
    const float* __restrict__ x, float* __restrict__ xn,
    const int* __restrict__ row, const int* __restrict__ col,
    const float* __restrict__ w, int nEdges) {
    __shared__ int   s_row[EDGE_TILE];
    __shared__ int   s_col[EDGE_TILE];
    __shared__ float s_w[EDGE_TILE];

    const int tid  = threadIdx.x;
    const int base = blockIdx.x * EDGE_TILE;

    // --- stage edge tile into LDS (async on CDNA5) -------------------------
    {
        const int sec  = tid >> 6;         // 0..3 : which array we stage
        const int lane = tid & 63;         // 0..63: which 16B chunk
        const int g    = base + lane * 4;  // first edge of this chunk
        // E is a multiple of 4, so 16B chunks never straddle the array end.
        if (sec < 3 && g < nEdges) {
            if (sec == 0)      tile_copy_b128(&row[g], &s_row[lane * 4]);
            else if (sec == 1) tile_copy_b128(&col[g], &s_col[lane * 4]);
            else               tile_copy_b128(&w[g],   &s_w[lane * 4]);
        }
    }
    wait_async_copies();   // per-wave ASYNCcnt drain
    __syncthreads();       // cross-wave visibility of LDS tile

    // --- consume: 16 lanes per edge, float4 per lane -----------------------
    const int grp = tid >> 4;   // which of the 16 in-flight edges
    const int sub = tid & 15;   // which float4 chunk of the 64-wide row

    #pragma unroll 4
    for (int j = 0; j < EDGE_TILE / EDGES_INFLIGHT; ++j) {
        const int el = j * EDGES_INFLIGHT + grp;
        const int e  = base + el;
        if (e < nEdges) {
            const int   r  = s_row[el];   // LDS broadcast across 16 lanes
            const int   c  = s_col[el];
            const float wt = s_w[el];

            const v4f_t v = *reinterpret_cast<const v4f_t*>(
                x + (size_t)r * DIM + (sub << 2));
            const v4f_t m = v * wt;

            float* dst = xn + (size_t)c * DIM + (sub << 2);
            atomicAdd(dst + 0, m.x);
            atomicAdd(dst + 1, m.y);
            atomicAdd(dst + 2, m.z);
            atomicAdd(dst + 3, m.w);
        }
    }
}

// ---------------------------------------------------------------------------
// init: acc = emb ; xa = emb ; xb = 0      (non-temporal b128 streams)
// ---------------------------------------------------------------------------
__global__ __launch_bounds__(TPB) void lightgcn_init(
    const v4f_t* __restrict__ emb, v4f_t* __restrict__ acc,
    v4f_t* __restrict__ xa, v4f_t* __restrict__ xb, int n4) {
    int i = blockIdx.x * TPB + threadIdx.x;
    if (i < n4) {
        const v4f_t e = __builtin_nontemporal_load(emb + i);
        const v4f_t z = {0.f, 0.f, 0.f, 0.f};
        __builtin_nontemporal_store(e, acc + i);
        __builtin_nontemporal_store(e, xa + i);
        __builtin_nontemporal_store(z, xb + i);
    }
}

// ---------------------------------------------------------------------------
// accum_zero: acc += xnew ; zbuf = 0   (zbuf becomes next layer's dest)
// ---------------------------------------------------------------------------
__global__ __launch_bounds__(TPB) void lightgcn_accum_zero(
    v4f_t* __restrict__ acc, const v4f_t* __restrict__ xnew,
    v4f_t* __restrict__ zbuf, int n4) {
    int i = blockIdx.x * TPB + threadIdx.x;
    if (i < n4) {
        const v4f_t a = __builtin_nontemporal_load(acc + i);
        const v4f_t b = __builtin_nontemporal_load(xnew + i);
        const v4f_t z = {0.f, 0.f, 0.f, 0.f};
        __builtin_nontemporal_store(a + b, acc + i);
        __builtin_nontemporal_store(z, zbuf + i);
    }
}

// ---------------------------------------------------------------------------
// finish: acc = (acc + xnew) * 0.25   (mean over layers 0..3)
// ---------------------------------------------------------------------------
__global__ __launch_bounds__(TPB) void lightgcn_finish(
    v4f_t* __restrict__ acc, const v4f_t* __restrict__ xnew, int n4) {
    int i = blockIdx.x * TPB + threadIdx.x;
    if (i < n4) {
        const v4f_t a = __builtin_nontemporal_load(acc + i);
        const v4f_t b = __builtin_nontemporal_load(xnew + i);
        __builtin_nontemporal_store((a + b) * 0.25f, acc + i);
    }
}

// ---------------------------------------------------------------------------
extern "C" void kernel_launch(void* const* d_in, const int* in_sizes, int n_in,
                              void* d_out, int out_size, void* d_ws, size_t ws_size,
                              hipStream_t stream) {
    const float* emb  = (const float*)d_in[0];   // [N, 64] f32
    const int*   eidx = (const int*)  d_in[1];   // [2, E] int32
    const float* ew   = (const float*)d_in[2];   // [E] f32

    const int E  = in_sizes[2];
    const int n  = in_sizes[0];      // N * DIM
    const int n4 = n / 4;

    const int* row = eidx;           // edge_index[0]
    const int* col = eidx + E;       // edge_index[1]

    float* acc = (float*)d_out;
    float* xa  = (float*)d_ws;                 // ping
    float* xb  = xa + (size_t)n;               // pong  (needs 2*n*4 bytes of ws)

    const dim3 blk(TPB);
    const int  gridN = (n4 + TPB - 1) / TPB;
    const int  gridE = (E + EDGE_TILE - 1) / EDGE_TILE;

    // acc = emb ; xa = emb ; xb = 0
    lightgcn_init<<<gridN, blk, 0, stream>>>(
        (const v4f_t*)emb, (v4f_t*)acc, (v4f_t*)xa, (v4f_t*)xb, n4);

    // layer 1: xb = A * xa ; acc += xb ; xa = 0
    lightgcn_scatter<<<gridE, blk, 0, stream>>>(xa, xb, row, col, ew, E);
    lightgcn_accum_zero<<<gridN, blk, 0, stream>>>(
        (v4f_t*)acc, (const v4f_t*)xb, (v4f_t*)xa, n4);

    // layer 2: xa = A * xb ; acc += xa ; xb = 0
    lightgcn_scatter<<<gridE, blk, 0, stream>>>(xb, xa, row, col, ew, E);
    lightgcn_accum_zero<<<gridN, blk, 0, stream>>>(
        (v4f_t*)acc, (const v4f_t*)xa, (v4f_t*)xb, n4);

    // layer 3: xb = A * xa ; acc = (acc + xb) / 4
    lightgcn_scatter<<<gridE, blk, 0, stream>>>(xa, xb, row, col, ew, E);
    lightgcn_finish<<<gridN, blk, 0, stream>>>(
        (v4f_t*)acc, (const v4f_t*)xb, n4);
}